// EdgeConv_59158879535425
// MI455X (gfx1250) — compile-verified
//
#include <hip/hip_runtime.h>

typedef __attribute__((ext_vector_type(2))) float v2f;
typedef __attribute__((ext_vector_type(8))) float v8f;

// Problem constants (match reference)
constexpr int BB  = 8;
constexpr int CC  = 64;      // input channels
constexpr int NN  = 2048;    // points
constexpr int KNB = 20;      // neighbors
constexpr int C2  = 128;     // 2*C edge-feature channels
constexpr int OC  = 64;      // out channels (both convs)
constexpr int NK  = NN * KNB;            // 40960 edge slots per batch

// Conv pipeline tiling
constexpr int TCOLS = 80;                 // edge-slot columns per workgroup (4 points * 20)
constexpr int TPTS  = TCOLS / KNB;        // 4 points per workgroup
constexpr int WGS_PER_BATCH = NK / TCOLS; // 512
constexpr int NWG   = BB * WGS_PER_BATCH; // 4096
constexpr int NTHREADS = 160;             // 5 waves, 16 cols each
constexpr int NWAVES   = 5;
constexpr int F1S = 132;                  // LDS col stride for 128-ch feat (bank-conflict pad)
constexpr int F2S = 68;                   // LDS col stride for 64-ch feat

// kNN tiling
constexpr int TQ     = 16;                // queries per workgroup
constexpr int CCHUNK = 128;               // candidates staged per chunk
constexpr int QWGS   = BB * (NN / TQ);    // 1024 workgroups
constexpr int ROWP   = 2052;              // dist row pitch (2052 % 64 == 4 -> conflict-free scans)
constexpr int PTS_S  = 68;                // transposed point tile stride

// ---------------------------------------------------------------------------
// Precompute per-point squared norms: sq[b][n] = sum_c P[b][c][n]^2
// ---------------------------------------------------------------------------
__global__ __launch_bounds__(256) void sq_kernel(const float* __restrict__ pts,
                                                 float* __restrict__ sqbuf)
{
    int i = blockIdx.x * 256 + threadIdx.x;
    if (i >= BB * NN) return;
    int bb = i / NN;
    int n  = i - bb * NN;
    const float* P = pts + (size_t)bb * CC * NN;
    float s = 0.f;
    for (int c = 0; c < CC; ++c) { float v = P[c * NN + n]; s += v * v; }
    sqbuf[i] = s;
}

// ---------------------------------------------------------------------------
// K1: kNN via WMMA Gram tiles. One workgroup per (batch, 16-query tile).
// dist[i][j] = sq_i - 2*<p_i,p_j> + sq_j held in (dynamic) LDS: 16 x 2052 f32.
// Candidates streamed in transposed 128-wide chunks; 8 waves each own a
// 16-candidate column slab (16x16x4 f32 WMMA, K=64 channels in 16 steps).
// Then 20x deterministic argmin per query (16 threads/query, u64 key tree).
// ---------------------------------------------------------------------------
__global__ __launch_bounds__(256) void knn_wmma_kernel(const float* __restrict__ pts,
                                                       const float* __restrict__ sqbuf,
                                                       int* __restrict__ idxout)
{
    extern __shared__ char dynsmem[];
    float* dist = (float*)dynsmem;                 // TQ * ROWP
    float* pT   = dist + TQ * ROWP;                // CCHUNK * PTS_S (cand-major, ch inner)
    float* qT   = pT + CCHUNK * PTS_S;             // TQ * PTS_S     (query-major, ch inner)
    float* sqi  = qT + TQ * PTS_S;                 // TQ
    unsigned long long* red = (unsigned long long*)(sqi + TQ);  // 256

    const int wg = blockIdx.x;
    const int bb = wg / (NN / TQ);
    const int q0 = (wg - bb * (NN / TQ)) * TQ;
    const int t    = threadIdx.x;
    const int lane = t & 31;
    const int wave = t >> 5;
    const float* P = pts + (size_t)bb * CC * NN;

    // Stage queries (transposed) + their norms
    for (int e = t; e < TQ * CC; e += 256) {
        int qi = e >> 6;
        int c  = e & 63;
        qT[qi * PTS_S + c] = P[c * NN + q0 + qi];
    }
    if (t < TQ) sqi[t] = sqbuf[(size_t)bb * NN + q0 + t];

    const int mrow = lane & 15;          // A row / D col within 16-tile
    const int koff = (lane >> 4) << 1;   // lanes 0-15 -> K0,K1; lanes 16-31 -> K2,K3

    for (int c0 = 0; c0 < NN; c0 += CCHUNK) {
        __syncthreads();                 // previous chunk's GEMM reads of pT done
        for (int e = t; e < CCHUNK * CC; e += 256) {
            int c = e >> 7;              // e / 128
            int m = e & (CCHUNK - 1);
            pT[m * PTS_S + c] = P[c * NN + c0 + m];
        }
        __syncthreads();

        v8f acc = {};
        for (int k4 = 0; k4 < CC; k4 += 4) {
            int kk = k4 + koff;
            v2f af = *(const v2f*)&qT[mrow * PTS_S + kk];
            v2f bf = *(const v2f*)&pT[((wave << 4) + mrow) * PTS_S + kk];
            acc = __builtin_amdgcn_wmma_f32_16x16x4_f32(
                false, af, false, bf, (short)0, acc, false, false);
        }

        int cand = c0 + (wave << 4) + mrow;
        float sj = sqbuf[(size_t)bb * NN + cand];
#pragma unroll
        for (int r = 0; r < 8; ++r) {
            int q = r + ((lane >> 4) << 3);
            dist[q * ROWP + cand] = sqi[q] - 2.f * acc[r] + sj;
        }
    }
    __syncthreads();

    // Selection: 16 threads per query, 20 deterministic argmins
    const int q  = t >> 4;
    const int s0 = t & 15;
    for (int it = 0; it < KNB; ++it) {
        unsigned long long best = ~0ull;
        for (int m = s0; m < NN; m += 16) {
            unsigned u = __float_as_uint(dist[q * ROWP + m]);
            u = (u & 0x80000000u) ? ~u : (u | 0x80000000u);  // monotonic map
            unsigned long long key = ((unsigned long long)u << 32) | (unsigned)m;
            if (key < best) best = key;
        }
        red[t] = best;
        __syncthreads();
        for (int s = 8; s > 0; s >>= 1) {
            if (s0 < s) {
                unsigned long long o = red[t + s];
                if (o < red[t]) red[t] = o;
            }
            __syncthreads();
        }
        if (s0 == 0) {
            int win = (int)(red[t] & 0xFFFFFFFFu);
            idxout[((size_t)bb * NN + q0 + q) * KNB + it] = win;
            dist[q * ROWP + win] = __uint_as_float(0x7F800000u);  // +inf
        }
        __syncthreads();
    }
}

// ---------------------------------------------------------------------------
// Conv pipeline. MODE 1: conv1 -> stats1 partials.
//                MODE 2: conv1+BN1+ReLU -> conv2 -> stats2 partials.
//                MODE 3: ... -> conv2+BN2+ReLU -> max over K -> out.
// ---------------------------------------------------------------------------
template<int MODE>
__global__ __launch_bounds__(NTHREADS) void conv_pipeline(
    const float* __restrict__ pts, const int* __restrict__ idxbuf,
    const float* __restrict__ W0g, const float* __restrict__ W1g,
    const float* __restrict__ ss1, const float* __restrict__ ss2,
    float* __restrict__ partials, float* __restrict__ out)
{
    __shared__ float smem[TCOLS * F1S];        // feat1, then overlaid feat2/out staging
    __shared__ float wsum[NWAVES * OC];
    __shared__ float wsq [NWAVES * OC];
    __shared__ float sc1[OC], sh1[OC], sc2[OC], sh2[OC];

    const int wg   = blockIdx.x;
    const int bb   = wg / WGS_PER_BATCH;
    const int tile = wg - bb * WGS_PER_BATCH;
    const int col0 = tile * TCOLS;
    const int n0   = col0 / KNB;
    const int t    = threadIdx.x;
    const int lane = t & 31;
    const int wave = t >> 5;

    if constexpr (MODE >= 2) { if (t < OC) { sc1[t] = ss1[t]; sh1[t] = ss1[OC + t]; } }
    if constexpr (MODE == 3) { if (t < OC) { sc2[t] = ss2[t]; sh2[t] = ss2[OC + t]; } }

    // Gather edge features: rows 0..63 = x_i, rows 64..127 = x_j - x_i (col-major LDS)
    const float* P = pts + (size_t)bb * CC * NN;
    const int*   I = idxbuf + (size_t)bb * NN * KNB;
    for (int e = t; e < TCOLS * CC; e += NTHREADS) {
        int c  = e / TCOLS;
        int cr = e - c * TCOLS;
        int pp = cr / KNB;
        int kk = cr - pp * KNB;
        int n  = n0 + pp;
        int j  = I[n * KNB + kk];
        float ctr = P[c * NN + n];
        float ngh = P[c * NN + j];
        smem[cr * F1S + c]      = ctr;
        smem[cr * F1S + CC + c] = ngh - ctr;
    }
    __syncthreads();

    // WMMA fragment lane mapping (16x16x4 f32)
    const int mrow = lane & 15;
    const int koff = (lane >> 4) << 1;
    const int lc   = (wave << 4) + mrow;   // this lane's column within the 80-col tile

    // ---- GEMM1: W0 (64x128) x feat (128 x 80) ----
    v8f acc[4] = {};
    for (int k4 = 0; k4 < C2; k4 += 4) {
        int kk = k4 + koff;
        v2f bf = *(const v2f*)&smem[lc * F1S + kk];
#pragma unroll
        for (int mt = 0; mt < 4; ++mt) {
            v2f af = *(const v2f*)&W0g[(mt * 16 + mrow) * C2 + kk];
            acc[mt] = __builtin_amdgcn_wmma_f32_16x16x4_f32(
                false, af, false, bf, (short)0, acc[mt], false, false);
        }
    }

    if constexpr (MODE == 1) {
#pragma unroll
        for (int mt = 0; mt < 4; ++mt) {
#pragma unroll
            for (int r = 0; r < 8; ++r) {
                float v = acc[mt][r];
                float s = v, qq = v * v;
#pragma unroll
                for (int m = 1; m < 16; m <<= 1) {
                    s  += __shfl_xor(s,  m, 32);
                    qq += __shfl_xor(qq, m, 32);
                }
                if (mrow == 0) {
                    int row = mt * 16 + r + ((lane >> 4) << 3);
                    wsum[wave * OC + row] = s;
                    wsq [wave * OC + row] = qq;
                }
            }
        }
        __syncthreads();
        if (t < OC) {
            float S = 0.f, Q = 0.f;
            for (int w = 0; w < NWAVES; ++w) { S += wsum[w * OC + t]; Q += wsq[w * OC + t]; }
            partials[(size_t)wg * (2 * OC) + t]      = S;
            partials[(size_t)wg * (2 * OC) + OC + t] = Q;
        }
        return;
    } else {
        // BN1 + ReLU -> stage as feat2 (col-major, stride F2S) overlaid on smem
        __syncthreads();
#pragma unroll
        for (int mt = 0; mt < 4; ++mt) {
#pragma unroll
            for (int r = 0; r < 8; ++r) {
                int row = mt * 16 + r + ((lane >> 4) << 3);
                float h = fmaxf(acc[mt][r] * sc1[row] + sh1[row], 0.f);
                smem[lc * F2S + row] = h;
            }
        }
        __syncthreads();

        // ---- GEMM2: W1 (64x64) x h1 (64 x 80) ----
        v8f acc2[4] = {};
        for (int k4 = 0; k4 < OC; k4 += 4) {
            int kk = k4 + koff;
            v2f bf = *(const v2f*)&smem[lc * F2S + kk];
#pragma unroll
            for (int mt = 0; mt < 4; ++mt) {
                v2f af = *(const v2f*)&W1g[(mt * 16 + mrow) * OC + kk];
                acc2[mt] = __builtin_amdgcn_wmma_f32_16x16x4_f32(
                    false, af, false, bf, (short)0, acc2[mt], false, false);
            }
        }

        if constexpr (MODE == 2) {
#pragma unroll
            for (int mt = 0; mt < 4; ++mt) {
#pragma unroll
                for (int r = 0; r < 8; ++r) {
                    float v = acc2[mt][r];
                    float s = v, qq = v * v;
#pragma unroll
                    for (int m = 1; m < 16; m <<= 1) {
                        s  += __shfl_xor(s,  m, 32);
                        qq += __shfl_xor(qq, m, 32);
                    }
                    if (mrow == 0) {
                        int row = mt * 16 + r + ((lane >> 4) << 3);
                        wsum[wave * OC + row] = s;
                        wsq [wave * OC + row] = qq;
                    }
                }
            }
            __syncthreads();
            if (t < OC) {
                float S = 0.f, Q = 0.f;
                for (int w = 0; w < NWAVES; ++w) { S += wsum[w * OC + t]; Q += wsq[w * OC + t]; }
                partials[(size_t)wg * (2 * OC) + t]      = S;
                partials[(size_t)wg * (2 * OC) + OC + t] = Q;
            }
        } else {
            // MODE 3: BN2 + ReLU -> stage, then max over K -> output
            __syncthreads();
#pragma unroll
            for (int mt = 0; mt < 4; ++mt) {
#pragma unroll
                for (int r = 0; r < 8; ++r) {
                    int row = mt * 16 + r + ((lane >> 4) << 3);
                    float h = fmaxf(acc2[mt][r] * sc2[row] + sh2[row], 0.f);
                    smem[lc * F2S + row] = h;
                }
            }
            __syncthreads();
            for (int e = t; e < TPTS * OC; e += NTHREADS) {
                int o = e & (OC - 1);
                int p = e >> 6;
                const float* base = &smem[(p * KNB) * F2S + o];
                float mx = base[0];
#pragma unroll
                for (int k2 = 1; k2 < KNB; ++k2) mx = fmaxf(mx, base[k2 * F2S]);
                out[((size_t)bb * OC + o) * NN + (n0 + p)] = mx;
            }
        }
    }
}

// ---------------------------------------------------------------------------
// Reduce per-WG partials -> folded BN scale/shift: y = x*scale + shift
// ---------------------------------------------------------------------------
__global__ void finalize_stats_kernel(const float* __restrict__ partials, int G,
                                      const float* __restrict__ gamma,
                                      const float* __restrict__ beta,
                                      float invM, float* __restrict__ ss)
{
    int ch = threadIdx.x;
    if (ch >= OC) return;
    float s = 0.f, q = 0.f;
    for (int g = 0; g < G; ++g) {
        s += partials[(size_t)g * (2 * OC) + ch];
        q += partials[(size_t)g * (2 * OC) + OC + ch];
    }
    float mean = s * invM;
    float var  = q * invM - mean * mean;
    float sc   = gamma[ch] * rsqrtf(var + 1e-5f);
    ss[ch]      = sc;
    ss[OC + ch] = beta[ch] - mean * sc;
}

// ---------------------------------------------------------------------------
extern "C" void kernel_launch(void* const* d_in, const int* in_sizes, int n_in,
                              void* d_out, int out_size, void* d_ws, size_t ws_size,
                              hipStream_t stream)
{
    (void)in_sizes; (void)n_in; (void)out_size; (void)ws_size;

    const float* points = (const float*)d_in[0];
    const float* W0     = (const float*)d_in[1];
    const float* g0     = (const float*)d_in[2];
    const float* b0     = (const float*)d_in[3];
    const float* W1     = (const float*)d_in[4];
    const float* g1     = (const float*)d_in[5];
    const float* b1     = (const float*)d_in[6];
    float* out = (float*)d_out;

    // Workspace layout (~5.6 MB): idx | part1 | part2 | ss1 | ss2 | sqbuf
    char* ws = (char*)d_ws;
    int* idxbuf = (int*)ws;
    size_t off = (size_t)BB * NN * KNB * sizeof(int);
    float* part1 = (float*)(ws + off);  off += (size_t)NWG * 2 * OC * sizeof(float);
    float* part2 = (float*)(ws + off);  off += (size_t)NWG * 2 * OC * sizeof(float);
    float* ss1   = (float*)(ws + off);  off += 2 * OC * sizeof(float);
    float* ss2   = (float*)(ws + off);  off += 2 * OC * sizeof(float);
    float* sqbuf = (float*)(ws + off);  off += (size_t)BB * NN * sizeof(float);

    const float invM = 1.0f / (float)((long long)BB * NN * KNB);

    // Dynamic LDS for kNN: dist(16x2052) + pT(128x68) + qT(16x68) + sqi(16) + red(256 u64)
    const size_t knn_lds = (size_t)(TQ * ROWP + CCHUNK * PTS_S + TQ * PTS_S + TQ) * sizeof(float)
                         + 256 * sizeof(unsigned long long);   // 172,608 B (<= 320KB/WGP)

    sq_kernel<<<(BB * NN + 255) / 256, 256, 0, stream>>>(points, sqbuf);
    knn_wmma_kernel<<<QWGS, 256, knn_lds, stream>>>(points, sqbuf, idxbuf);
    conv_pipeline<1><<<NWG, NTHREADS, 0, stream>>>(points, idxbuf, W0, W1,
                                                   nullptr, nullptr, part1, nullptr);
    finalize_stats_kernel<<<1, 64, 0, stream>>>(part1, NWG, g0, b0, invM, ss1);
    conv_pipeline<2><<<NWG, NTHREADS, 0, stream>>>(points, idxbuf, W0, W1,
                                                   ss1, nullptr, part2, nullptr);
    finalize_stats_kernel<<<1, 64, 0, stream>>>(part2, NWG, g1, b1, invM, ss2);
    conv_pipeline<3><<<NWG, NTHREADS, 0, stream>>>(points, idxbuf, W0, W1,
                                                   ss1, ss2, nullptr, out);
}